// ChamferLoss_8117488189452
// MI455X (gfx1250) — compile-verified
//
#include <hip/hip_runtime.h>

typedef __attribute__((ext_vector_type(2))) float v2f;
typedef __attribute__((ext_vector_type(8))) float v8f;

#define BATCH 4
#define NPTS 8192
#define TILES (NPTS / 16)        // 512 column tiles
#define WAVES_PER_BLOCK 8
#define TILES_PER_WAVE 2         // two A row-tiles per wave share each B tile
#define BLOCKS_PER_DIR (TILES / (WAVES_PER_BLOCK * TILES_PER_WAVE))  // 32

// Kernel 1: per (batch, direction, pair of 16-row tiles) compute min squared
// distance from each "x" point to all NPTS "y" points using f32 WMMA.
__global__ __launch_bounds__(256) void chamfer_min_kernel(
    const float* __restrict__ pred,
    const float* __restrict__ gt,
    float* __restrict__ minsq /* [2][BATCH][NPTS] */)
{
    extern __shared__ float4 lds4[];  // NPTS float4 = 128KB: [y0,y1,y2,||y||^2]

    const int g   = blockIdx.x;
    const int tg  = g % BLOCKS_PER_DIR;
    const int dir = (g / BLOCKS_PER_DIR) & 1;
    const int b   = g / (BLOCKS_PER_DIR * 2);

    const float* x = (dir == 0) ? pred : gt;   // rows (min taken per row)
    const float* y = (dir == 0) ? gt   : pred; // columns
    x += (size_t)b * NPTS * 3;
    y += (size_t)b * NPTS * 3;
    float* out = minsq + ((size_t)dir * BATCH + b) * NPTS;

    // Stage target cloud into LDS with precomputed norms.
    for (int i = threadIdx.x; i < NPTS; i += 256) {
        const float y0 = y[3 * i + 0];
        const float y1 = y[3 * i + 1];
        const float y2 = y[3 * i + 2];
        lds4[i] = make_float4(y0, y1, y2, y0 * y0 + y1 * y1 + y2 * y2);
    }
    __syncthreads();

    const int wave = threadIdx.x >> 5;
    const int lane = threadIdx.x & 31;
    const bool hi  = lane >= 16;
    const int tile0 = (tg * WAVES_PER_BLOCK + wave) * TILES_PER_WAVE;
    const int i0 = tile0 * 16;        // first row tile
    const int i1 = i0 + 16;           // second row tile

    // A tile: row m = [-2x0, -2x1, -2x2, ||x||^2]
    // A 16x4 f32 layout: VGPR0 = K0 (lanes 0-15) / K2 (lanes 16-31),
    //                    VGPR1 = K1 / K3; M = lane & 15.
    const int m = lane & 15;
    v2f a0, a1;
    {
        const float* xp = x + (size_t)(i0 + m) * 3;
        const float x0 = xp[0], x1 = xp[1], x2 = xp[2];
        const float xn = x0 * x0 + x1 * x1 + x2 * x2;
        a0.x = hi ? (-2.0f * x2) : (-2.0f * x0);
        a0.y = hi ? xn           : (-2.0f * x1);
    }
    {
        const float* xp = x + (size_t)(i1 + m) * 3;
        const float x0 = xp[0], x1 = xp[1], x2 = xp[2];
        const float xn = x0 * x0 + x1 * x1 + x2 * x2;
        a1.x = hi ? (-2.0f * x2) : (-2.0f * x0);
        a1.y = hi ? xn           : (-2.0f * x1);
    }

    v8f acc0, acc1;
#pragma unroll
    for (int v = 0; v < 8; ++v) { acc0[v] = 3.0e38f; acc1[v] = 3.0e38f; }

    for (int t = 0; t < TILES; ++t) {
        // B col n = [y0, y1, y2, 1]; layout mirrors A; N = lane & 15.
        const float4 q = lds4[t * 16 + (lane & 15)];
        v2f bv;
        bv.x = hi ? q.z  : q.x;
        bv.y = hi ? 1.0f : q.y;
        // C[v][lane] = ||y(N)||^2 so WMMA yields d = ||x||^2 + ||y||^2 - 2 x.y
        v8f c;
#pragma unroll
        for (int v = 0; v < 8; ++v) c[v] = q.w;
        v8f d0 = __builtin_amdgcn_wmma_f32_16x16x4_f32(
            false, a0, false, bv, (short)0, c, false, false);
        v8f d1 = __builtin_amdgcn_wmma_f32_16x16x4_f32(
            false, a1, false, bv, (short)0, c, false, false);
#pragma unroll
        for (int v = 0; v < 8; ++v) {
            acc0[v] = fminf(acc0[v], d0[v]);
            acc1[v] = fminf(acc1[v], d1[v]);
        }
    }

    // Row-wise min: butterfly across each 16-lane group (wave32).
#pragma unroll
    for (int mask = 1; mask <= 8; mask <<= 1) {
#pragma unroll
        for (int v = 0; v < 8; ++v) {
            acc0[v] = fminf(acc0[v], __shfl_xor(acc0[v], mask, 32));
            acc1[v] = fminf(acc1[v], __shfl_xor(acc1[v], mask, 32));
        }
    }
    // D layout: lanes 0-15 carry rows M=v, lanes 16-31 carry rows M=v+8.
    if (lane == 0) {
#pragma unroll
        for (int v = 0; v < 8; ++v) {
            out[i0 + v] = acc0[v];
            out[i1 + v] = acc1[v];
        }
    } else if (lane == 16) {
#pragma unroll
        for (int v = 0; v < 8; ++v) {
            out[i0 + 8 + v] = acc0[v];
            out[i1 + 8 + v] = acc1[v];
        }
    }
}

// Kernel 2: reduce [2][BATCH][NPTS] min distances to the scalar loss:
//   loss_b = mean(pred2gt_b) + mean(gt2pred_b) + max(pred2gt_b)
//   out    = mean_b(loss_b)
__global__ __launch_bounds__(1024) void chamfer_reduce_kernel(
    const float* __restrict__ minsq, float* __restrict__ out)
{
    __shared__ float ssum[1024];
    __shared__ float smax[1024];
    const int tid = threadIdx.x;
    float total = 0.0f;
    for (int b = 0; b < BATCH; ++b) {
        const float* p2g = minsq + (size_t)b * NPTS;            // dir 0
        const float* g2p = minsq + (size_t)(BATCH + b) * NPTS;  // dir 1
        float s = 0.0f, mx = -3.0e38f;
        for (int i = tid; i < NPTS; i += 1024) {
            const float v = p2g[i];
            s += v + g2p[i];
            mx = fmaxf(mx, v);
        }
        ssum[tid] = s;
        smax[tid] = mx;
        __syncthreads();
        for (int st = 512; st > 0; st >>= 1) {
            if (tid < st) {
                ssum[tid] += ssum[tid + st];
                smax[tid] = fmaxf(smax[tid], smax[tid + st]);
            }
            __syncthreads();
        }
        if (tid == 0) total += ssum[0] * (1.0f / NPTS) + smax[0];
        __syncthreads();
    }
    if (tid == 0) out[0] = total * (1.0f / BATCH);
}

extern "C" void kernel_launch(void* const* d_in, const int* in_sizes, int n_in,
                              void* d_out, int out_size, void* d_ws, size_t ws_size,
                              hipStream_t stream) {
    (void)in_sizes; (void)n_in; (void)out_size; (void)ws_size;
    const float* pred = (const float*)d_in[0];
    const float* gt   = (const float*)d_in[1];
    float* minsq = (float*)d_ws;  // 2 * BATCH * NPTS floats = 256KB

    dim3 grid(BATCH * 2 * BLOCKS_PER_DIR);  // 256 blocks, 8 waves each
    chamfer_min_kernel<<<grid, 256, NPTS * sizeof(float4), stream>>>(
        pred, gt, minsq);
    chamfer_reduce_kernel<<<1, 1024, 0, stream>>>(minsq, (float*)d_out);
}